// Adaptive_Spectral_Block1_71863392796960
// MI455X (gfx1250) — compile-verified
//
#include <hip/hip_runtime.h>
#include <hip/hip_bf16.h>

// ---------------------------------------------------------------------------
// Problem constants (from reference): B=8, C=64, N=1024, T=24
// ---------------------------------------------------------------------------
#define BB 8
#define CC 64
#define NN 1024
#define TT 24
#define CT (CC * TT)      // 1536
#define KSEL 819          // int(1024 * 0.8)

typedef float v2f __attribute__((ext_vector_type(2)));
typedef float v8f __attribute__((ext_vector_type(8)));

// ---------------------------------------------------------------------------
// CDNA5 async global->LDS copy (ASYNCcnt path), per CDNA5 ISA:
//   GV mode: copies 16B per lane from a 64-bit global address to the LDS offset in VDST.
// LDS destination operand is a VGPR holding the LDS byte offset; the low 32
// bits of a flat __shared__ pointer are exactly that offset (aperture rule).
// ---------------------------------------------------------------------------
__device__ __forceinline__ void async_load16(unsigned lds_off, const void* gptr) {
    asm volatile("global_load_async_to_lds_b128 %0, %1, off"
                 :
                 : "v"(lds_off), "v"((unsigned long long)(uintptr_t)gptr)
                 : "memory");
}
__device__ __forceinline__ void wait_async0() {
    asm volatile("s_wait_asynccnt 0x0" ::: "memory");
}

// ---------------------------------------------------------------------------
// Block-wide reduction helpers (256 threads)
// ---------------------------------------------------------------------------
__device__ __forceinline__ float blk_max(float v, float* red, int tid) {
    red[tid] = v; __syncthreads();
    #pragma unroll
    for (int s = 128; s >= 1; s >>= 1) {
        if (tid < s) red[tid] = fmaxf(red[tid], red[tid + s]);
        __syncthreads();
    }
    float r = red[0]; __syncthreads();
    return r;
}

__device__ __forceinline__ float blk_sum(float v, float* red, int tid) {
    red[tid] = v; __syncthreads();
    #pragma unroll
    for (int s = 128; s >= 1; s >>= 1) {
        if (tid < s) red[tid] += red[tid + s];
        __syncthreads();
    }
    float r = red[0]; __syncthreads();
    return r;
}

__device__ __forceinline__ int blk_isum(int v, int* red, int tid) {
    red[tid] = v; __syncthreads();
    #pragma unroll
    for (int s = 128; s >= 1; s >>= 1) {
        if (tid < s) red[tid] += red[tid + s];
        __syncthreads();
    }
    int r = red[0]; __syncthreads();
    return r;
}

// ---------------------------------------------------------------------------
// Kernel 1: h[b,t,n,c] = sum_c' x[b,c',n,t] * W[c',c]
// Writes hIn[b][n][c*T+t]  (GEMM-A friendly layout, coalesced writes)
// and    hsum[b][n][c] = sum_t h
// Grid: (N, B), block: 256
// ---------------------------------------------------------------------------
__global__ void h_kernel(const float* __restrict__ x, const float* __restrict__ W,
                         float* __restrict__ hIn, float* __restrict__ hsum) {
    const int n = blockIdx.x, b = blockIdx.y, tid = threadIdx.x;
    __shared__ float xslice[CC][TT];         // 64 x 24 = 6 KB
    __shared__ float part[4][CC];

    const size_t xbase = ((size_t)b * CC) * NN * TT + (size_t)n * TT;
    for (int i = tid; i < CC * TT; i += 256) {
        int cp = i / TT, t = i % TT;
        xslice[cp][t] = x[xbase + (size_t)cp * NN * TT + t];
    }
    __syncthreads();

    const int c = tid & 63;          // output channel
    const int tg = tid >> 6;         // t-group 0..3 (6 t's each)
    float acc[6] = {0.f, 0.f, 0.f, 0.f, 0.f, 0.f};
    for (int cp = 0; cp < CC; ++cp) {
        float w = W[cp * CC + c];
        #pragma unroll
        for (int j = 0; j < 6; ++j)
            acc[j] = fmaf(w, xslice[cp][tg * 6 + j], acc[j]);
    }

    float* hout = hIn + ((size_t)b * NN + n) * CT;
    float s = 0.f;
    #pragma unroll
    for (int j = 0; j < 6; ++j) {
        hout[c * TT + tg * 6 + j] = acc[j];
        s += acc[j];
    }
    part[tg][c] = s;
    __syncthreads();
    if (tid < CC) {
        float hs = part[0][tid] + part[1][tid] + part[2][tid] + part[3][tid];
        hsum[((size_t)b * NN + n) * CC + tid] = hs;
    }
}

// ---------------------------------------------------------------------------
// Kernel 2: Wh1[b,n] = h_ . a[:C],  Wh2[b,n] = h_ . a[C:]
// Grid: 32 blocks x 256 threads (8192 rows)
// ---------------------------------------------------------------------------
__global__ void wh_kernel(const float* __restrict__ hsum, const float* __restrict__ a,
                          float* __restrict__ wh1, float* __restrict__ wh2) {
    const int idx = blockIdx.x * 256 + threadIdx.x;   // b*N + n
    const float* hr = hsum + (size_t)idx * CC;
    float s1 = 0.f, s2 = 0.f;
    for (int c = 0; c < CC; ++c) {
        float hv = hr[c];
        s1 = fmaf(hv, a[c], s1);
        s2 = fmaf(hv, a[CC + c], s2);
    }
    wh1[idx] = s1;
    wh2[idx] = s2;
}

// ---------------------------------------------------------------------------
// Kernel 3: Z[b][n][m] = relu(scale * dot(A[b][n][:], B'[m][:]))  (K = 64)
// A = hsum (per-batch), B' row-major [1024][64]; b_stride_B = 0 for `memory`,
// N*C for per-batch h_.   WMMA f32 16x16x4.  Block tile 64x64, 8 waves.
// Tile staged with async global->LDS (ASYNCcnt).
// Grid: (16, 16, 8), block: 256
// ---------------------------------------------------------------------------
__global__ void gemm_nt_relu_kernel(const float* __restrict__ Abase,
                                    const float* __restrict__ Bbase,
                                    int b_stride_B, float scale,
                                    float* __restrict__ Out) {
    const int bm = blockIdx.x, bn = blockIdx.y, b = blockIdx.z;
    const int tid = threadIdx.x;
    const float* A  = Abase + (size_t)b * NN * CC;
    const float* Bp = Bbase + (size_t)b * b_stride_B;
    __shared__ float As[64][64];   // [rowLocal][k]
    __shared__ float Bs[64][64];   // [colLocal][k]

    const unsigned asBase = (unsigned)(uintptr_t)&As[0][0];
    const unsigned bsBase = (unsigned)(uintptr_t)&Bs[0][0];
    #pragma unroll
    for (int j = 0; j < 4; ++j) {              // 4096 floats per tile, f4/lane
        const int e = tid * 4 + j * 1024;      // 16B-aligned linear element
        const int r = e >> 6, k = e & 63;
        async_load16(asBase + e * 4, &A [(size_t)(bm * 64 + r) * CC + k]);
        async_load16(bsBase + e * 4, &Bp[(size_t)(bn * 64 + r) * CC + k]);
    }
    wait_async0();
    __syncthreads();

    const int w = tid >> 5, lane = tid & 31;
    const int wr = w & 3, wc = w >> 2;              // 4x2 wave grid
    const int mrow  = 16 * wr + (lane & 15);
    const int kb    = (lane >> 4) * 2;              // lanes 16-31 hold K+2,K+3
    const int colL0 = 32 * wc + (lane & 15);

    v8f acc0 = {}; v8f acc1 = {};
    #pragma unroll
    for (int ks = 0; ks < 16; ++ks) {
        const int k0 = ks * 4 + kb;
        v2f av; av.x = As[mrow][k0];       av.y = As[mrow][k0 + 1];
        v2f b0; b0.x = Bs[colL0][k0];      b0.y = Bs[colL0][k0 + 1];
        v2f b1; b1.x = Bs[colL0 + 16][k0]; b1.y = Bs[colL0 + 16][k0 + 1];
        acc0 = __builtin_amdgcn_wmma_f32_16x16x4_f32(false, av, false, b0,
                                                     (short)0, acc0, false, false);
        acc1 = __builtin_amdgcn_wmma_f32_16x16x4_f32(false, av, false, b1,
                                                     (short)0, acc1, false, false);
    }

    float* out = Out + (size_t)b * NN * NN;
    const int hi = (lane >> 4) << 3;                // D: lanes>=16 hold M=r+8
    const int col0 = bn * 64 + 32 * wc + (lane & 15);
    #pragma unroll
    for (int r = 0; r < 8; ++r) {
        int row = bm * 64 + 16 * wr + r + hi;
        out[(size_t)row * NN + col0]      = fmaxf(acc0[r] * scale, 0.0f);
        out[(size_t)row * NN + col0 + 16] = fmaxf(acc1[r] * scale, 0.0f);
    }
}

// ---------------------------------------------------------------------------
// Kernel 4: per (b, n) row:
//   A1 = softmax(Z1 row); A2 = softmax(Z2 row)
//   F  = relu(cwa*A1 + cw*A2)          (softmax(relu(F)) is order-preserving,
//                                       so top-k is taken directly on F)
//   exact top-k(819) threshold incl. jax.lax.top_k tie order (ascending index)
//   att = softmax( sel ? leaky_relu(Wh1[n]+Wh2[m]) : -inf )
// att overwrites Z1 in place.   Grid: (N, B), block 256, 4 elements/thread.
// ---------------------------------------------------------------------------
__global__ void attention_rows_kernel(float* z1_att, const float* __restrict__ Z2,
                                      const float* __restrict__ cwa,
                                      const float* __restrict__ cw,
                                      const float* __restrict__ wh1,
                                      const float* __restrict__ wh2) {
    const int n = blockIdx.x, b = blockIdx.y, tid = threadIdx.x;
    __shared__ float red[256];
    __shared__ int  ired[256];

    float* z1row = z1_att + ((size_t)b * NN + n) * NN;
    const float* z2row = Z2 + ((size_t)b * NN + n) * NN;
    const int m0 = tid * 4;

    // -- softmax of Z1 row, weighted by cwa ---------------------------------
    float zv[4], fv[4];
    float mx = -1e30f;
    #pragma unroll
    for (int j = 0; j < 4; ++j) { zv[j] = z1row[m0 + j]; mx = fmaxf(mx, zv[j]); }
    mx = blk_max(mx, red, tid);
    float s = 0.f;
    #pragma unroll
    for (int j = 0; j < 4; ++j) s += expf(zv[j] - mx);
    s = blk_sum(s, red, tid);
    float inv = 1.0f / s;
    #pragma unroll
    for (int j = 0; j < 4; ++j)
        fv[j] = cwa[(size_t)n * NN + m0 + j] * (expf(zv[j] - mx) * inv);

    // -- softmax of Z2 row, weighted by cw, combine + relu ------------------
    mx = -1e30f;
    #pragma unroll
    for (int j = 0; j < 4; ++j) { zv[j] = z2row[m0 + j]; mx = fmaxf(mx, zv[j]); }
    mx = blk_max(mx, red, tid);
    s = 0.f;
    #pragma unroll
    for (int j = 0; j < 4; ++j) s += expf(zv[j] - mx);
    s = blk_sum(s, red, tid);
    inv = 1.0f / s;
    #pragma unroll
    for (int j = 0; j < 4; ++j)
        fv[j] = fmaxf(fv[j] + cw[(size_t)n * NN + m0 + j] * (expf(zv[j] - mx) * inv), 0.0f);

    // -- exact k-th largest via binary search on (non-negative) float bits --
    unsigned bits[4];
    #pragma unroll
    for (int j = 0; j < 4; ++j) bits[j] = __float_as_uint(fv[j]);
    unsigned lo = 0u, hi = 0xFFFFFFFFu;
    while (lo < hi) {                     // uniform: driven by block reduction
        unsigned mid = lo + ((hi - lo) >> 1);
        int c = 0;
        #pragma unroll
        for (int j = 0; j < 4; ++j) c += (bits[j] > mid) ? 1 : 0;
        int tot = blk_isum(c, ired, tid);
        if (tot < KSEL) hi = mid; else lo = mid + 1u;
    }
    const unsigned v = lo;                // k-th largest value (as bits)

    int g = 0, ecnt = 0, eq[4];
    #pragma unroll
    for (int j = 0; j < 4; ++j) {
        g += (bits[j] > v) ? 1 : 0;
        eq[j] = (bits[j] == v) ? 1 : 0;
        ecnt += eq[j];
    }
    const int need = KSEL - blk_isum(g, ired, tid);

    // exclusive prefix over per-thread tie counts (ascending index order)
    ired[tid] = ecnt; __syncthreads();
    for (int off = 1; off < 256; off <<= 1) {
        int t = (tid >= off) ? ired[tid - off] : 0;
        __syncthreads();
        ired[tid] += t;
        __syncthreads();
    }
    int rank = ired[tid] - ecnt;
    __syncthreads();

    int sel[4];
    #pragma unroll
    for (int j = 0; j < 4; ++j) {
        if (bits[j] > v)      sel[j] = 1;
        else if (eq[j])     { sel[j] = (rank < need) ? 1 : 0; rank++; }
        else                  sel[j] = 0;
    }

    // -- masked leaky-relu attention softmax --------------------------------
    const float w1 = wh1[b * NN + n];
    float ev[4];
    float mxa = -1e30f;
    #pragma unroll
    for (int j = 0; j < 4; ++j) {
        float e  = w1 + wh2[b * NN + m0 + j];
        float lr = (e > 0.f) ? e : 0.01f * e;
        ev[j] = sel[j] ? lr : -1e30f;
        mxa = fmaxf(mxa, ev[j]);
    }
    mxa = blk_max(mxa, red, tid);
    float sa = 0.f;
    #pragma unroll
    for (int j = 0; j < 4; ++j) {
        ev[j] = (ev[j] < -1e29f) ? 0.0f : expf(ev[j] - mxa);
        sa += ev[j];
    }
    sa = blk_sum(sa, red, tid);
    const float inva = 1.0f / sa;
    #pragma unroll
    for (int j = 0; j < 4; ++j) z1row[m0 + j] = ev[j] * inva;
}

// ---------------------------------------------------------------------------
// Kernel 5: out[b][c][m][t] = relu( sum_n hIn[b][n][ct] * att[b][n][m] )
// Per-batch GEMM: M = ct (1536), N' = m (1024), K = n (1024).
// WMMA f32 16x16x4, 64x64 block tile.  K staged in 32-chunks through a
// 2-deep LDS double buffer filled with async global->LDS loads so the next
// chunk's fetch overlaps the current chunk's WMMAs (wait only after compute).
// Grid: (24, 16, 8), block: 256
// ---------------------------------------------------------------------------
__global__ void gemm_final_kernel(const float* __restrict__ hIn,
                                  const float* __restrict__ att,
                                  float* __restrict__ out) {
    const int bm = blockIdx.x, bn = blockIdx.y, b = blockIdx.z;
    const int tid = threadIdx.x;
    __shared__ float As[2][32][64];   // [buf][kLocal][ctLocal]  (8 KB each)
    __shared__ float Bs[2][32][64];   // [buf][kLocal][mLocal]
    const float* Ab = hIn + (size_t)b * NN * CT;
    const float* Bb = att + (size_t)b * NN * NN;

    const unsigned asBase = (unsigned)(uintptr_t)&As[0][0][0];
    const unsigned bsBase = (unsigned)(uintptr_t)&Bs[0][0][0];

    // issue async fill of buffer `buf` with K-chunk `kc` (2048 floats/array)
    auto issue_chunk = [&](int buf, int kc) {
        const int k0g = kc * 32;
        #pragma unroll
        for (int j = 0; j < 2; ++j) {
            const int e = tid * 4 + j * 1024;      // 16B-aligned linear element
            const int kL = e >> 6, cl = e & 63;
            async_load16(asBase + buf * 8192 + e * 4,
                         &Ab[(size_t)(k0g + kL) * CT + bm * 64 + cl]);
            async_load16(bsBase + buf * 8192 + e * 4,
                         &Bb[(size_t)(k0g + kL) * NN + bn * 64 + cl]);
        }
    };

    const int w = tid >> 5, lane = tid & 31;
    const int wr = w & 3, wc = w >> 2;
    const int mrowL = 16 * wr + (lane & 15);
    const int kb    = (lane >> 4) * 2;
    const int colL0 = 32 * wc + (lane & 15);

    issue_chunk(0, 0);
    wait_async0();
    __syncthreads();

    v8f acc0 = {}; v8f acc1 = {};
    for (int kc = 0; kc < 32; ++kc) {
        const int cur = kc & 1;
        if (kc + 1 < 32) issue_chunk(cur ^ 1, kc + 1);   // prefetch next chunk
        #pragma unroll
        for (int ks = 0; ks < 8; ++ks) {
            const int k0 = ks * 4 + kb;
            v2f av; av.x = As[cur][k0][mrowL];      av.y = As[cur][k0 + 1][mrowL];
            v2f b0; b0.x = Bs[cur][k0][colL0];      b0.y = Bs[cur][k0 + 1][colL0];
            v2f b1; b1.x = Bs[cur][k0][colL0 + 16]; b1.y = Bs[cur][k0 + 1][colL0 + 16];
            acc0 = __builtin_amdgcn_wmma_f32_16x16x4_f32(false, av, false, b0,
                                                         (short)0, acc0, false, false);
            acc1 = __builtin_amdgcn_wmma_f32_16x16x4_f32(false, av, false, b1,
                                                         (short)0, acc1, false, false);
        }
        wait_async0();      // prefetch done (issued before compute, so it overlapped)
        __syncthreads();    // all waves finished reading `cur` before it is refilled
    }

    const int hi = (lane >> 4) << 3;
    const int mIdx0 = bn * 64 + 32 * wc + (lane & 15);
    #pragma unroll
    for (int r = 0; r < 8; ++r) {
        int ct = bm * 64 + 16 * wr + r + hi;
        int c = ct / TT, t = ct % TT;
        size_t base = (((size_t)b * CC + c) * NN) * TT + t;
        out[base + (size_t)mIdx0 * TT]        = fmaxf(acc0[r], 0.0f);
        out[base + (size_t)(mIdx0 + 16) * TT] = fmaxf(acc1[r], 0.0f);
    }
}

// ---------------------------------------------------------------------------
// Host-side launch
// ---------------------------------------------------------------------------
extern "C" void kernel_launch(void* const* d_in, const int* in_sizes, int n_in,
                              void* d_out, int out_size, void* d_ws, size_t ws_size,
                              hipStream_t stream) {
    (void)in_sizes; (void)n_in; (void)out_size;
    const float* x      = (const float*)d_in[0];  // [B,C,N,T]
    const float* W      = (const float*)d_in[1];  // [C,C]
    const float* a      = (const float*)d_in[2];  // [2C,1]
    const float* memory = (const float*)d_in[3];  // [N,C]
    const float* cwa    = (const float*)d_in[4];  // [N,N]
    const float* cw     = (const float*)d_in[5];  // [N,N]
    float* out = (float*)d_out;                   // [B,C,N,T]
    float* ws  = (float*)d_ws;

    size_t off = 0;
    float* hsum = ws + off; off += (size_t)BB * NN * CC;       //  2 MB
    float* hIn  = ws + off; off += (size_t)BB * NN * CT;       // 48 MB
    float* Z1   = ws + off; off += (size_t)BB * NN * NN;       // 32 MB (reused as att)
    float* Z2   = ws + off; off += (size_t)BB * NN * NN;       // 32 MB
    float* wh1  = ws + off; off += (size_t)BB * NN;
    float* wh2  = ws + off; off += (size_t)BB * NN;
    if (ws_size < off * sizeof(float)) return;   // workspace too small: bail safely

    const float scale = 0.125f;                  // 1/sqrt(C)

    h_kernel<<<dim3(NN, BB), 256, 0, stream>>>(x, W, hIn, hsum);
    wh_kernel<<<dim3((BB * NN) / 256), 256, 0, stream>>>(hsum, a, wh1, wh2);
    gemm_nt_relu_kernel<<<dim3(16, 16, BB), 256, 0, stream>>>(hsum, memory, 0, scale, Z1);
    gemm_nt_relu_kernel<<<dim3(16, 16, BB), 256, 0, stream>>>(hsum, hsum, NN * CC, scale, Z2);
    attention_rows_kernel<<<dim3(NN, BB), 256, 0, stream>>>(Z1, Z2, cwa, cw, wh1, wh2);
    gemm_final_kernel<<<dim3(CT / 64, NN / 64, BB), 256, 0, stream>>>(hIn, Z1, out);
}